// LocalUpsampleConvLayer_20976620274392
// MI455X (gfx1250) — compile-verified
//
#include <hip/hip_runtime.h>

// CDNA5 / gfx1250 WMMA vector types
typedef __attribute__((ext_vector_type(2))) float v2f;
typedef __attribute__((ext_vector_type(8))) float v8f;

#define Bn    4
#define ICn   32
#define OCn   32
#define OHn   128
#define OWn   128
#define QV    (ICn * 3 * 3)   // 288 = reduction length per pixel
#define PITCH (QV + 4)        // 292: bank-conflict-free LDS pitch
#define WPP   (OCn * QV)      // 9216 weight floats per output pixel (36.9 KB)

__global__ __launch_bounds__(256) void luconv_wmma_kernel(
    const float* __restrict__ x,      // [4][32][64][64]
    const float* __restrict__ weight, // [128][128][32][32][3][3]
    const float* __restrict__ bias,   // [32][128][128]
    float* __restrict__ out)          // [4][32][128][128]
{
  __shared__ float sA[8 * Bn * PITCH]; // per-wave A (patches): 8 waves x 4 x 292
  __shared__ float sO[128 * 9];        // output restage: 128 (b*32+oc) rows x 8 w (pitch 9)

  const int lane = threadIdx.x & 31;
  const int wv   = threadIdx.x >> 5;          // wave in block: 0..7
  const int pos  = blockIdx.x * 8 + wv;       // one output pixel per wave
  const int h    = pos >> 7;
  const int w    = pos & 127;

  const float* wpos = weight + (size_t)pos * WPP;

  // Prefetch the whole 36KB per-pixel weight block (288 x 128B lines) so the
  // HBM stream overlaps the patch gather (emits global_prefetch_b8).
  for (int i = lane; i < 288; i += 32)
    __builtin_prefetch((const char*)wpos + (size_t)i * 128, 0, 3);

  // ---- Build A (patches) in LDS: rows = batch, cols = q = ic*9 + ki*3 + kj ----
  // upsample x2 (nearest) + reflect pad 1, all folded into the gather index.
  float* pA = &sA[wv * (Bn * PITCH)];
  for (int e = lane; e < Bn * QV; e += 32) {   // 1152 / 32 = 36 uniform iters
    int b  = e / QV;
    int q  = e - b * QV;
    int ic = q / 9;
    int r  = q - ic * 9;
    int ki = r / 3;
    int kj = r - ki * 3;
    int uh = h + ki - 1;
    uh = (uh < 0) ? -uh : ((uh >= OHn) ? (2 * OHn - 2 - uh) : uh);
    int uw = w + kj - 1;
    uw = (uw < 0) ? -uw : ((uw >= OWn) ? (2 * OWn - 2 - uw) : uw);
    pA[b * PITCH + q] = x[((b * ICn + ic) * 64 + (uh >> 1)) * 64 + (uw >> 1)];
  }
  __syncthreads();

  // ---- WMMA main loop: out[4x32] = A[4x288] @ W[288x32], K in chunks of 4 ----
  // A layout (f32 16x4): lanes 0-15 M=0..15 K={k0,k0+1}; lanes 16-31 K={k0+2,k0+3}.
  // Rows >=4 are don't-care (we replicate batch rows via lane&3; result rows unused).
  // B layout (f32 4x16): lane n = column (oc); same K split -> one b64 per lane.
  const int nrow  = lane & 15;     // oc within tile / D column
  const int khalf = lane >> 4;     // 0 -> K=k0,k0+1 ; 1 -> K=k0+2,k0+3
  const float* bptr0 = wpos + nrow * QV + khalf * 2;   // oc tile 0..15
  const float* bptr1 = bptr0 + 16 * QV;                // oc tile 16..31
  const float* aptr  = pA + (lane & 3) * PITCH + khalf * 2;

  v8f acc0 = {0.f, 0.f, 0.f, 0.f, 0.f, 0.f, 0.f, 0.f};
  v8f acc1 = acc0;

  #pragma unroll 4
  for (int k0 = 0; k0 < QV; k0 += 4) {          // 72 iters, 2 WMMA each
    v2f a  = *(const v2f*)(aptr);               // ds_load_b64 (broadcast, no conflicts)
    v2f b0 = *(const v2f*)(bptr0);              // global_load_b64: fresh weights
    v2f b1 = *(const v2f*)(bptr1);
    acc0 = __builtin_amdgcn_wmma_f32_16x16x4_f32(
        false, a, false, b0, (short)0, acc0, false, false);
    acc1 = __builtin_amdgcn_wmma_f32_16x16x4_f32(
        false, a, false, b1, (short)0, acc1, false, false);
    aptr  += 4;
    bptr0 += 4;
    bptr1 += 4;
  }

  // ---- Epilogue: bias add + restage through LDS for coalesced stores ----
  // D layout: lanes 0-15 hold M=0..7 in acc[0..7]; rows 0..3 are the batch.
  if (lane < 16) {
    float bv0 = bias[nrow * (OHn * OWn) + pos];
    float bv1 = bias[(nrow + 16) * (OHn * OWn) + pos];
    #pragma unroll
    for (int b = 0; b < Bn; ++b) {
      sO[(b * OCn + nrow) * 9 + wv]        = acc0[b] + bv0;
      sO[(b * OCn + 16 + nrow) * 9 + wv]   = acc1[b] + bv1;
    }
  }
  __syncthreads();

  // Block covers 8 consecutive w positions (same h row): store 8x4B contiguous
  // runs per (b,oc) row -> sector-aligned coalesced writes of the 8MB output.
  const int pos_base = blockIdx.x * 8;
  const int w8    = threadIdx.x & 7;
  const int rbase = threadIdx.x >> 3;   // 0..31
  #pragma unroll
  for (int pass = 0; pass < 4; ++pass) {
    int r = pass * 32 + rbase;          // r = b*32 + oc, 0..127
    out[(size_t)r * (OHn * OWn) + pos_base + w8] = sO[r * 9 + w8];
  }
}

extern "C" void kernel_launch(void* const* d_in, const int* in_sizes, int n_in,
                              void* d_out, int out_size, void* d_ws, size_t ws_size,
                              hipStream_t stream) {
  const float* x    = (const float*)d_in[0];
  const float* wgt  = (const float*)d_in[1];
  const float* bias = (const float*)d_in[2];
  float* out        = (float*)d_out;

  dim3 grid((OHn * OWn) / 8);  // 2048 blocks, 8 waves each -> one wave per pixel
  dim3 block(256);
  luconv_wmma_kernel<<<grid, block, 0, stream>>>(x, wgt, bias, out);
}